// Attention_82660940579436
// MI455X (gfx1250) — compile-verified
//
#include <hip/hip_runtime.h>

typedef float v2f __attribute__((ext_vector_type(2)));
typedef float v8f __attribute__((ext_vector_type(8)));
typedef int   v4i __attribute__((__vector_size__(4 * sizeof(int))));

#define S_   4096
#define B_   32
#define H_   512
#define ROWS 64          // s-rows per block in fused kernel
#define NC   (S_/ROWS)   // 64 chunks per batch
#define KC   64          // k-chunk staged in LDS
#define NCH  (H_/KC)     // 8 k-chunks
#define SC1  16          // s-chunks for mean partials (256 rows each)

#if __has_builtin(__builtin_amdgcn_global_load_async_to_lds_b128) && \
    __has_builtin(__builtin_amdgcn_s_wait_asynccnt)
#define USE_ASYNC 1
#else
#define USE_ASYNC 0
#endif

#if __has_builtin(__builtin_amdgcn_tanhf)
#define TANHF(x) __builtin_amdgcn_tanhf(x)
#else
#define TANHF(x) tanhf(x)
#endif

// ---------------------------------------------------------------------------
// K0: pack Wk[H][H] (row-major, out x in) into WkP[kpair][n][2] so that the
// WMMA B-matrix fetch (per-lane K-pair at column n) is one coalesced 8B load.
__global__ void pack_wk(const float* __restrict__ Wk, float* __restrict__ WkP) {
    const int kp = blockIdx.x;        // 0..255  (pair index: k = 2*kp, 2*kp+1)
    const int t  = threadIdx.x;       // 0..255
    for (int i = 0; i < 2; ++i) {
        const int n = t + i * 256;    // output-feature column 0..511
        v2f v;
        v.x = Wk[(size_t)n * H_ + 2 * kp];
        v.y = Wk[(size_t)n * H_ + 2 * kp + 1];
        *(v2f*)(WkP + ((size_t)kp * H_ + n) * 2) = v;
    }
}

// ---------------------------------------------------------------------------
// K1: partial sums over S for the time-mean of hidden_states.
__global__ void mean_partial(const float* __restrict__ hs, float* __restrict__ part) {
    const int sc = blockIdx.x, b = blockIdx.y, t = threadIdx.x;
    float a0 = 0.f, a1 = 0.f;
    for (int s = sc * 256; s < sc * 256 + 256; ++s) {
        const float* row = hs + ((size_t)s * B_ + b) * H_;
        a0 += row[t];
        a1 += row[t + 256];
    }
    float* p = part + ((size_t)sc * B_ + b) * H_;
    p[t]       = a0;
    p[t + 256] = a1;
}

// ---------------------------------------------------------------------------
// K2: hbar = sum(partials)/S;  q[b][h] = hbar . Wq[h,:] + bq[h]
__global__ void q_proj(const float* __restrict__ part, const float* __restrict__ Wq,
                       const float* __restrict__ bq, float* __restrict__ q) {
    const int b = blockIdx.x, t = threadIdx.x;
    __shared__ float hbar[H_];
    for (int i = 0; i < 2; ++i) {
        const int h = t + i * 256;
        float a = 0.f;
        for (int sc = 0; sc < SC1; ++sc) a += part[((size_t)sc * B_ + b) * H_ + h];
        hbar[h] = a * (1.0f / S_);
    }
    __syncthreads();
    for (int i = 0; i < 2; ++i) {
        const int h = t + i * 256;
        float a = bq[h];
        const float* wr = Wq + (size_t)h * H_;
        for (int k = 0; k < H_; ++k) a += hbar[k] * wr[k];
        q[(size_t)b * H_ + h] = a;
    }
}

// ---------------------------------------------------------------------------
// K3: fused key projection (f32 WMMA) + tanh + scores + online-softmax chunk
// partials. grid (NC, B), 256 threads = 8 waves. Wave w owns output columns
// [w*64, w*64+64); all waves share the 64 s-rows of this chunk.
// h-tiles are staged global->LDS with CDNA5 async-DMA, double-buffered.
__global__ void __launch_bounds__(256)
fused_attn(const float* __restrict__ hs, const float* __restrict__ WkP,
           const float* __restrict__ bk, const float* __restrict__ q,
           const long long* __restrict__ lengths,
           float* __restrict__ accOut, float* __restrict__ mlOut) {
    const int c = blockIdx.x, b = blockIdx.y;
    const int s0   = c * ROWS;
    const int tid  = threadIdx.x;
    const int lane = tid & 31, w = tid >> 5;
    const int half = lane >> 4, l16 = lane & 15;
    const int n0   = w * 64;

    __shared__ __align__(16) float hT[2][ROWS][KC + 4]; // +4 pad: conflict-free A reads
    __shared__ float scoreW[8][ROWS];
    __shared__ float pL[ROWS];
    __shared__ float mlL[2];

    v8f acc[4][4] = {};                  // 4 m-tiles x 4 n-tiles = 64x64 block

    float qv[4], bkv[4];
#pragma unroll
    for (int nt = 0; nt < 4; ++nt) {
        const int col = n0 + nt * 16 + l16;
        qv[nt]  = q[(size_t)b * H_ + col];
        bkv[nt] = bk[col];
    }

    const int lrow = tid >> 2;           // 0..63
    const int lseg = (tid & 3) * 16;     // 0,16,32,48
    const float* srcBase = hs + ((size_t)(s0 + lrow) * B_ + b) * H_ + lseg;

    // 16 k-steps of a 64-wide k-chunk: 16 WMMAs per step from buffer bufi
    auto computeChunk = [&](int bufi, int k0) {
        for (int kk = 0; kk < KC; kk += 4) {
            const int kg = k0 + kk;
            v2f Bm[4], Am[4];
#pragma unroll
            for (int nt = 0; nt < 4; ++nt) {
                const int col = n0 + nt * 16 + l16;
                // lanes 0-15: K=kg,kg+1 ; lanes 16-31: K=kg+2,kg+3 (B layout)
                Bm[nt] = *(const v2f*)(WkP + ((size_t)((kg >> 1) + half) * H_ + col) * 2);
            }
#pragma unroll
            for (int mt = 0; mt < 4; ++mt)
                Am[mt] = *(const v2f*)&hT[bufi][mt * 16 + l16][kk + 2 * half];
#pragma unroll
            for (int mt = 0; mt < 4; ++mt)
#pragma unroll
                for (int nt = 0; nt < 4; ++nt)
                    acc[mt][nt] = __builtin_amdgcn_wmma_f32_16x16x4_f32(
                        false, Am[mt], false, Bm[nt], (short)0, acc[mt][nt],
                        false, false);
        }
    };

#if USE_ASYNC
    // per-thread 16 contiguous floats per chunk = 4 async B128 copies; the
    // immediate offset is added to BOTH the global and LDS addresses (ISA).
    auto issueCopy = [&](int bufi, int k0) {
        auto g = (__attribute__((address_space(1))) v4i*)(srcBase + k0);
        auto l = (__attribute__((address_space(3))) v4i*)&hT[bufi][lrow][lseg];
        __builtin_amdgcn_global_load_async_to_lds_b128(g, l, 0,  0);
        __builtin_amdgcn_global_load_async_to_lds_b128(g, l, 16, 0);
        __builtin_amdgcn_global_load_async_to_lds_b128(g, l, 32, 0);
        __builtin_amdgcn_global_load_async_to_lds_b128(g, l, 48, 0);
    };

    issueCopy(0, 0);
#pragma unroll 1
    for (int i = 0; i < NCH - 1; ++i) {
        issueCopy((i + 1) & 1, (i + 1) * KC);          // prefetch next chunk
        __builtin_amdgcn_s_wait_asynccnt(4);           // chunk i landed (in-order)
        __syncthreads();                               // visible to all waves
        computeChunk(i & 1, i * KC);
        __syncthreads();                               // buffer free for reuse
    }
    __builtin_amdgcn_s_wait_asynccnt(0);
    __syncthreads();
    computeChunk((NCH - 1) & 1, (NCH - 1) * KC);
#else
    for (int k0 = 0; k0 < H_; k0 += KC) {
        float4 r0 = *(const float4*)(srcBase + k0 + 0);
        float4 r1 = *(const float4*)(srcBase + k0 + 4);
        float4 r2 = *(const float4*)(srcBase + k0 + 8);
        float4 r3 = *(const float4*)(srcBase + k0 + 12);
        __syncthreads();
        *(float4*)&hT[0][lrow][lseg + 0]  = r0;
        *(float4*)&hT[0][lrow][lseg + 4]  = r1;
        *(float4*)&hT[0][lrow][lseg + 8]  = r2;
        *(float4*)&hT[0][lrow][lseg + 12] = r3;
        __syncthreads();
        computeChunk(0, k0);
    }
#endif

    // bias + tanh -> key values stay resident in the accumulator registers
#pragma unroll
    for (int mt = 0; mt < 4; ++mt)
#pragma unroll
        for (int nt = 0; nt < 4; ++nt)
#pragma unroll
            for (int v = 0; v < 8; ++v)
                acc[mt][nt][v] = TANHF(acc[mt][nt][v] + bkv[nt]);

    // per-wave score partials: row r = mt*16 + v + 8*half, cols = this wave's 64
#pragma unroll
    for (int mt = 0; mt < 4; ++mt) {
#pragma unroll
        for (int v = 0; v < 8; ++v) {
            float sp = 0.f;
#pragma unroll
            for (int nt = 0; nt < 4; ++nt) sp += acc[mt][nt][v] * qv[nt];
            sp += __shfl_xor(sp, 1, 32);
            sp += __shfl_xor(sp, 2, 32);
            sp += __shfl_xor(sp, 4, 32);
            sp += __shfl_xor(sp, 8, 32);
            if (l16 == 0) scoreW[w][mt * 16 + v + 8 * half] = sp;
        }
    }
    __syncthreads();

    // fixed-order cross-wave reduction + length mask (additive -10000)
    const long long len = lengths[b];
    if (tid < ROWS) {
        float sv = 0.f;
        for (int ww = 0; ww < 8; ++ww) sv += scoreW[ww][tid];
        if ((long long)(s0 + tid) >= len) sv -= 10000.0f;
        scoreW[0][tid] = sv;   // each thread touches only its own column: no race
    }
    __syncthreads();

    // chunk softmax stats (single thread, fixed order => deterministic)
    if (tid == 0) {
        float m = -3.0e38f;
        for (int r = 0; r < ROWS; ++r) m = fmaxf(m, scoreW[0][r]);
        float l = 0.f;
        for (int r = 0; r < ROWS; ++r) {
            const float pp = __expf(scoreW[0][r] - m);
            pL[r] = pp;
            l += pp;
        }
        mlL[0] = m;
        mlL[1] = l;
    }
    __syncthreads();

    // weighted key accumulation from resident registers: acc_out[n] = sum_r p[r]*key[r][n]
    float* aOut = accOut + ((size_t)b * NC + c) * H_;
#pragma unroll
    for (int nt = 0; nt < 4; ++nt) {
        float part = 0.f;
#pragma unroll
        for (int mt = 0; mt < 4; ++mt)
#pragma unroll
            for (int v = 0; v < 8; ++v)
                part += pL[mt * 16 + v + 8 * half] * acc[mt][nt][v];
        part += __shfl_xor(part, 16, 32);   // fold rows r and r+8 halves
        if (half == 0) aOut[n0 + nt * 16 + l16] = part;
    }
    if (tid == 0) {
        mlOut[((size_t)b * NC + c) * 2 + 0] = mlL[0];
        mlOut[((size_t)b * NC + c) * 2 + 1] = mlL[1];
    }
}

// ---------------------------------------------------------------------------
// K4: deterministic flash-combine of the NC chunk partials per batch.
__global__ void combine(const float* __restrict__ accP, const float* __restrict__ mlP,
                        float* __restrict__ out) {
    const int b = blockIdx.x, t = threadIdx.x;
    __shared__ float wc[NC];
    __shared__ float Linv;
    if (t == 0) {
        float M = -3.0e38f;
        for (int c2 = 0; c2 < NC; ++c2) M = fmaxf(M, mlP[((size_t)b * NC + c2) * 2]);
        float L = 0.f;
        for (int c2 = 0; c2 < NC; ++c2) {
            const float e = __expf(mlP[((size_t)b * NC + c2) * 2] - M);
            wc[c2] = e;
            L += e * mlP[((size_t)b * NC + c2) * 2 + 1];
        }
        Linv = 1.0f / L;
    }
    __syncthreads();
    for (int i = 0; i < 2; ++i) {
        const int h = t + i * 256;
        float o = 0.f;
        for (int c2 = 0; c2 < NC; ++c2)
            o += wc[c2] * accP[((size_t)b * NC + c2) * H_ + h];
        out[(size_t)b * H_ + h] = o * Linv;
    }
}

// ---------------------------------------------------------------------------
extern "C" void kernel_launch(void* const* d_in, const int* in_sizes, int n_in,
                              void* d_out, int out_size, void* d_ws, size_t ws_size,
                              hipStream_t stream) {
    const float*     hs      = (const float*)d_in[0];
    const float*     Wq      = (const float*)d_in[1];
    const float*     bq      = (const float*)d_in[2];
    const float*     Wk      = (const float*)d_in[3];
    const float*     bk      = (const float*)d_in[4];
    const long long* lengths = (const long long*)d_in[5];   // jnp.int64
    float* out = (float*)d_out;

    float* ws    = (float*)d_ws;
    float* WkP   = ws;                 // 512*512        = 262144 floats
    float* mpart = WkP   + 262144;     // 16*32*512      = 262144
    float* qbuf  = mpart + 262144;     // 32*512         = 16384
    float* accP  = qbuf  + 16384;      // 32*64*512      = 1048576
    float* mlP   = accP  + 1048576;    // 32*64*2        = 4096   (~6.1 MB total)

    pack_wk     <<<dim3(256),      dim3(256), 0, stream>>>(Wk, WkP);
    mean_partial<<<dim3(SC1, B_),  dim3(256), 0, stream>>>(hs, mpart);
    q_proj      <<<dim3(B_),       dim3(256), 0, stream>>>(mpart, Wq, bq, qbuf);
    fused_attn  <<<dim3(NC, B_),   dim3(256), 0, stream>>>(hs, WkP, bk, qbuf, lengths, accP, mlP);
    combine     <<<dim3(B_),       dim3(256), 0, stream>>>(accP, mlP, out);
}